// Aggregator_59803124629826
// MI455X (gfx1250) — compile-verified
//
#include <hip/hip_runtime.h>
#include <math.h>

#define DDIM 64

typedef float v2f __attribute__((ext_vector_type(2)));
typedef float v8f __attribute__((ext_vector_type(8)));

// ---------------------------------------------------------------------------
// 0) zero output + counts
// ---------------------------------------------------------------------------
__global__ void zero_kernel(float* __restrict__ out, long n_out,
                            float* __restrict__ cnt, int n_ent) {
    long total = n_out + (long)n_ent;
    for (long i = (long)blockIdx.x * blockDim.x + threadIdx.x; i < total;
         i += (long)gridDim.x * blockDim.x) {
        if (i < n_out) out[i] = 0.0f;
        else           cnt[i - n_out] = 0.0f;
    }
}

// ---------------------------------------------------------------------------
// 1) disen_weight = softmax(disen_weight_att, -1) @ relation_emb   [F x 64]
//    one block, 64 threads (thread t = output column t)
// ---------------------------------------------------------------------------
__global__ void disen_kernel(const float* __restrict__ att,   // [F, R]
                             const float* __restrict__ rel,   // [R, 64]
                             float* __restrict__ dw,          // [F, 64]
                             int n_fac, int n_rel) {
    __shared__ float sw[4][32];
    int t = threadIdx.x;
    if (t < n_fac) {
        float mx = -1e30f;
        for (int r = 0; r < n_rel; ++r) mx = fmaxf(mx, att[t * n_rel + r]);
        float s = 0.0f;
        for (int r = 0; r < n_rel; ++r) {
            float e = expf(att[t * n_rel + r] - mx);
            sw[t][r] = e; s += e;
        }
        float inv = 1.0f / s;
        for (int r = 0; r < n_rel; ++r) sw[t][r] *= inv;
    }
    __syncthreads();
    for (int f = 0; f < n_fac; ++f) {
        float acc = 0.0f;
        for (int r = 0; r < n_rel; ++r) acc += sw[f][r] * rel[r * DDIM + t];
        dw[f * DDIM + t] = acc;
    }
}

// ---------------------------------------------------------------------------
// 2) KG aggregate: scatter entity_emb[tail]*relation_emb[etype] into seg_sum
//    one thread per (edge, 16B chunk); entity_emb (25.6MB) is L2-resident
// ---------------------------------------------------------------------------
__global__ void edge_kernel(const float* __restrict__ ent,
                            const float* __restrict__ rel,
                            const int* __restrict__ head,
                            const int* __restrict__ tail,
                            const int* __restrict__ etype,
                            float* __restrict__ seg_sum,
                            float* __restrict__ cnt, int n_edges) {
    int idx = blockIdx.x * blockDim.x + threadIdx.x;
    int total = n_edges * 16;
    if (idx >= total) return;
    int e = idx >> 4, ch = idx & 15;
    int h = head[e], t = tail[e], r = etype[e];
    float4 ev = *(const float4*)(ent + (size_t)t * DDIM + ch * 4);
    float4 rv = *(const float4*)(rel + (size_t)r * DDIM + ch * 4);
    float* dst = seg_sum + (size_t)h * DDIM + ch * 4;
    unsafeAtomicAdd(dst + 0, ev.x * rv.x);
    unsafeAtomicAdd(dst + 1, ev.y * rv.y);
    unsafeAtomicAdd(dst + 2, ev.z * rv.z);
    unsafeAtomicAdd(dst + 3, ev.w * rv.w);
    if (ch == 0) unsafeAtomicAdd(cnt + h, 1.0f);
}

// ---------------------------------------------------------------------------
// 3) entity_agg = seg_sum / max(cnt, 1)
// ---------------------------------------------------------------------------
__global__ void finalize_kernel(float* __restrict__ seg_sum,
                                const float* __restrict__ cnt, int n_ent) {
    int i = blockIdx.x * blockDim.x + threadIdx.x;
    int total = n_ent * DDIM;
    if (i >= total) return;
    float c = cnt[i >> 6];
    seg_sum[i] = seg_sum[i] / fmaxf(c, 1.0f);
}

// ---------------------------------------------------------------------------
// 4) COO SPMM: user_agg += interact_vals * entity_emb[cols], scatter by rows
// ---------------------------------------------------------------------------
__global__ void spmm_kernel(const float* __restrict__ ent,
                            const float* __restrict__ vals,
                            const int* __restrict__ rows,
                            const int* __restrict__ cols,
                            float* __restrict__ user_agg, int nnz) {
    int idx = blockIdx.x * blockDim.x + threadIdx.x;
    int total = nnz * 16;
    if (idx >= total) return;
    int e = idx >> 4, ch = idx & 15;
    int r = rows[e], c = cols[e];
    float v = vals[e];
    float4 ev = *(const float4*)(ent + (size_t)c * DDIM + ch * 4);
    float* dst = user_agg + (size_t)r * DDIM + ch * 4;
    unsafeAtomicAdd(dst + 0, ev.x * v);
    unsafeAtomicAdd(dst + 1, ev.y * v);
    unsafeAtomicAdd(dst + 2, ev.z * v);
    unsafeAtomicAdd(dst + 3, ev.w * v);
}

// ---------------------------------------------------------------------------
// 5) User fuse with WMMA (fp32 16x16x4):
//    score = softmax(user_emb @ latent_emb^T)           (GEMM1, N padded to 16)
//    user_agg = (score @ disen_weight) * user_agg + user_agg   (GEMM2, fused)
//    One wave = one 16-user tile.
// ---------------------------------------------------------------------------
__global__ void user_fuse_kernel(const float* __restrict__ user_emb, // [U,64]
                                 const float* __restrict__ latent,   // [F,64]
                                 const float* __restrict__ dw,       // [F,64]
                                 float* __restrict__ user_out,       // [U,64] in/out
                                 int n_fac, int n_tiles) {
    __shared__ float s_score[8][16][4];               // per-wave scratch
    const int wave = threadIdx.x >> 5;
    const int lane = threadIdx.x & 31;
    const int tile = blockIdx.x * 8 + wave;
    if (tile >= n_tiles) return;                      // whole-wave exit, EXEC stays all-1
    const int row0 = tile * 16;
    const int half = lane >> 4;                       // 0 or 1
    const int l15  = lane & 15;
    const bool nvalid = (l15 < n_fac);                // only first F columns real
    const float bmask = nvalid ? 1.0f : 0.0f;         // zero-pad B cols >= F
    const int   safe_n = nvalid ? l15 : 0;            // in-range row for dead lanes

    // ---- GEMM1: raw = user_emb[tile] @ latent^T, K=64 via 16x (16x16x4) ----
    // A(16x4): lane half h -> VGPR0 = K=2h, VGPR1 = K=2h+1, row = l15
    // B(4x16): same K mapping, col = l15; dead lanes load row 0 * 0.0f
    const float* arow = user_emb + (size_t)(row0 + l15) * DDIM + 2 * half;
    const float* brow = latent   + (size_t)safe_n * DDIM + 2 * half;
    v8f c = {0.f, 0.f, 0.f, 0.f, 0.f, 0.f, 0.f, 0.f};
    #pragma unroll
    for (int k0 = 0; k0 < DDIM; k0 += 4) {
        v2f a; a.x = arow[k0];         a.y = arow[k0 + 1];
        v2f b; b.x = brow[k0] * bmask; b.y = brow[k0 + 1] * bmask;
        c = __builtin_amdgcn_wmma_f32_16x16x4_f32(false, a, false, b,
                                                  (short)0, c, false, false);
    }

    // ---- softmax over n=0..3 (aligned 4-lane groups; C: VGPR r, lane 16h+n
    //      holds raw[m=r+8h][n]) ----
    float sc[8];
    #pragma unroll
    for (int r = 0; r < 8; ++r) {
        float v  = c[r];
        float mx = v;
        mx = fmaxf(mx, __shfl_xor(mx, 1, 32));
        mx = fmaxf(mx, __shfl_xor(mx, 2, 32));
        float e = nvalid ? expf(v - mx) : 0.0f;
        float s = e;
        s += __shfl_xor(s, 1, 32);
        s += __shfl_xor(s, 2, 32);
        sc[r] = nvalid ? (e / s) : 0.0f;
    }

    // ---- C-layout -> A-layout via per-wave LDS (wave-local sync) ----
    if (nvalid) {
        #pragma unroll
        for (int r = 0; r < 8; ++r)
            s_score[wave][r + 8 * half][l15] = sc[r];
    }
    asm volatile("s_wait_dscnt 0" ::: "memory");
    v2f a2;
    a2.x = s_score[wave][l15][2 * half + 0];
    a2.y = s_score[wave][l15][2 * half + 1];

    // ---- GEMM2: boost = score @ disen_weight, four 16-col blocks; fuse ----
    #pragma unroll
    for (int n0 = 0; n0 < DDIM; n0 += 16) {
        v2f b2;
        b2.x = dw[(2 * half + 0) * DDIM + n0 + l15];
        b2.y = dw[(2 * half + 1) * DDIM + n0 + l15];
        v8f d = {0.f, 0.f, 0.f, 0.f, 0.f, 0.f, 0.f, 0.f};
        d = __builtin_amdgcn_wmma_f32_16x16x4_f32(false, a2, false, b2,
                                                  (short)0, d, false, false);
        const int col = n0 + l15;
        #pragma unroll
        for (int r = 0; r < 8; ++r) {
            const int m = r + 8 * half;
            const size_t idx = (size_t)(row0 + m) * DDIM + col;
            float ua = user_out[idx];
            user_out[idx] = ua * (1.0f + d[r]);   // (score@dw)*ua + ua
        }
    }
}

// ---------------------------------------------------------------------------
extern "C" void kernel_launch(void* const* d_in, const int* in_sizes, int n_in,
                              void* d_out, int out_size, void* d_ws, size_t ws_size,
                              hipStream_t stream) {
    const float* entity_emb  = (const float*)d_in[0];
    const float* user_emb    = (const float*)d_in[1];
    const float* latent_emb  = (const float*)d_in[2];
    const float* relation    = (const float*)d_in[3];
    const float* disen_att   = (const float*)d_in[4];
    const float* ivals       = (const float*)d_in[5];
    const int*   edge_index  = (const int*)d_in[6];
    const int*   edge_type   = (const int*)d_in[7];
    const int*   irows       = (const int*)d_in[8];
    const int*   icols       = (const int*)d_in[9];

    const int n_ent   = in_sizes[0] / DDIM;     // 100000
    const int n_usr   = in_sizes[1] / DDIM;     // 50000
    const int n_fac   = in_sizes[2] / DDIM;     // 4
    const int n_rel   = in_sizes[3] / DDIM;     // 32
    const int nnz     = in_sizes[5];            // 1M
    const int n_edges = in_sizes[7];            // 1M

    const int* head = edge_index;               // edge_index[0]
    const int* tail = edge_index + n_edges;     // edge_index[1]

    float* entity_agg = (float*)d_out;                            // [n_ent,64]
    float* user_agg   = (float*)d_out + (size_t)n_ent * DDIM;     // [n_usr,64]

    float* cnt = (float*)d_ws;                  // [n_ent]
    float* dw  = cnt + n_ent;                   // [n_fac,64]

    const int B = 256;
    long n_out = (long)out_size;

    // 0) zero outputs + counts
    {
        long total = n_out + n_ent;
        int blocks = (int)((total + B - 1) / B);
        zero_kernel<<<blocks, B, 0, stream>>>(entity_agg, n_out, cnt, n_ent);
    }
    // 1) disen_weight (tiny)
    disen_kernel<<<1, DDIM, 0, stream>>>(disen_att, relation, dw, n_fac, n_rel);
    // 2) KG scatter
    {
        int total = n_edges * 16;
        edge_kernel<<<(total + B - 1) / B, B, 0, stream>>>(
            entity_emb, relation, head, tail, edge_type, entity_agg, cnt, n_edges);
    }
    // 3) segment mean finalize
    {
        int total = n_ent * DDIM;
        finalize_kernel<<<(total + B - 1) / B, B, 0, stream>>>(entity_agg, cnt, n_ent);
    }
    // 4) COO SPMM scatter into user_agg
    {
        int total = nnz * 16;
        spmm_kernel<<<(total + B - 1) / B, B, 0, stream>>>(
            entity_emb, ivals, irows, icols, user_agg, nnz);
    }
    // 5) WMMA user fuse (in-place on user_agg)
    {
        int n_tiles = (n_usr + 15) / 16;        // 3125
        int blocks  = (n_tiles + 7) / 8;        // 8 waves/block, 1 tile/wave
        user_fuse_kernel<<<blocks, 256, 0, stream>>>(
            user_emb, latent_emb, dw, user_agg, n_fac, n_tiles);
    }
}